// RRF2d_32710470926513
// MI455X (gfx1250) — compile-verified
//
#include <hip/hip_runtime.h>

// RRF2d: locally-connected 3x3 conv, fp32.
//   x: (32,32,64,64)  w: (64,288,4096)  b: (64,4096)  out: (32,64,4096)
// Per location p: GEMM M=32(batch) x N=64(Cout) x K=288 via
// V_WMMA_F32_16X16X4_F32 (exact fp32). HBM-bound on the 302MB weight
// stream -> weights double-buffered through LDS with async-to-LDS so the
// stream overlaps the WMMA pipeline; each weight byte read from HBM once
// and reused across the batch (x32) in registers.

#define HH    64
#define WW    64
#define CI    32
#define CO    64
#define LTOT  4096
#define KTOT  288            // CI * 9
#define KC    16             // K-chunk staged in LDS
#define NCHK  (KTOT / KC)    // 18 chunks
#define PBLK  8              // spatial locations per workgroup (1 per wave)

typedef __attribute__((ext_vector_type(2))) float v2f;
typedef __attribute__((ext_vector_type(4))) float v4f;
typedef __attribute__((ext_vector_type(8))) float v8f;
typedef __attribute__((ext_vector_type(2))) int   v2i;
typedef __attribute__((ext_vector_type(4))) int   v4i;

typedef __attribute__((address_space(1))) v4i v4i_glob;
typedef __attribute__((address_space(3))) v4i v4i_lds;

#ifndef __has_builtin
#define __has_builtin(x) 0
#endif

#if __has_builtin(__builtin_amdgcn_global_load_async_to_lds_b128)
#define USE_ASYNC_LDS 1
#else
#define USE_ASYNC_LDS 0
#endif

__global__ __launch_bounds__(256) void rrf2d_wmma_kernel(
    const float* __restrict__ xin,   // (32,32,64,64)
    const float* __restrict__ wgt,   // (64,288,4096)
    const float* __restrict__ bias,  // (64,4096)
    float* __restrict__ out)         // (32,64,4096)
{
    const int tid  = threadIdx.x;
    const int wave = tid >> 5;
    const int lane = tid & 31;
    const int half = lane >> 4;
    const int l16  = lane & 15;

    const int p0 = blockIdx.x * PBLK;   // 512 blocks cover L=4096
    const int p  = p0 + wave;           // this wave's location
    const int y0 = p >> 6;
    const int x0 = p & 63;

    // double-buffered weight chunk [buf][l][o][p]: 2*16*64*8*4B = 64 KB
    __shared__ float lds_w[2][KC][CO][PBLK];
    // K-index -> packed {(c*4096 + (kh-1)*64 + (kw-1) + 65) << 4 | tap}
    __shared__ int tab[KTOT];

    if (tid < KTOT) {
        const int c  = tid / 9;
        const int r9 = tid - c * 9;
        const int kh = r9 / 3;
        const int kw = r9 - kh * 3;
        tab[tid] = ((c * (HH * WW) + (kh - 1) * WW + (kw - 1) + 65) << 4) | r9;
    }

    // per-wave 9-bit tap validity mask (boundary / zero-pad handling)
    int vmask = 0;
    #pragma unroll
    for (int r = 0; r < 9; ++r) {
        const int yy = y0 + (r / 3) - 1;
        const int xx = x0 + (r % 3) - 1;
        vmask |= (int)((unsigned)yy < (unsigned)HH && (unsigned)xx < (unsigned)WW) << r;
    }

    // x address collapses to b*131072 + p + dOff[kg]
    const float* abase[2];
    abase[0] = xin + (size_t)l16 * (CI * HH * WW) + p - 65;
    abase[1] = abase[0] + (size_t)16 * (CI * HH * WW);

    // ---- coalesced async stage of W[o, k0..k0+KC-1, p0..p0+7] into buf ----
    auto stage = [&](int chunk, int buf) {
        const int k0 = chunk * KC;
        #pragma unroll
        for (int pass = 0; pass < 8; ++pass) {
            const int pair = pass * 128 + (tid >> 1);   // (l,o) pairs
            const int lloc = pair >> 6;
            const int o    = pair & 63;
            const int poff = (tid & 1) * 4;
            const float* gsrc =
                wgt + ((size_t)o * KTOT + (size_t)(k0 + lloc)) * LTOT + (p0 + poff);
            float* ldst = &lds_w[buf][lloc][o][poff];
#if USE_ASYNC_LDS
            __builtin_amdgcn_global_load_async_to_lds_b128(
                (v4i_glob*)(__attribute__((address_space(1))) void*)(void*)gsrc,
                (v4i_lds*)(__attribute__((address_space(3))) void*)(void*)ldst,
                0, 0);
#else
            *(v4f*)ldst = *(const v4f*)gsrc;
#endif
        }
    };

    stage(0, 0);   // prologue prefetch

    v8f acc[2][4] = {};   // 2 M-tiles (batch) x 4 N-tiles (Cout)

    for (int ch = 0; ch < NCHK; ++ch) {
        const int buf = ch & 1;
#if USE_ASYNC_LDS
  #if __has_builtin(__builtin_amdgcn_s_wait_asynccnt)
        __builtin_amdgcn_s_wait_asynccnt(0);
  #else
        asm volatile("s_wait_asynccnt 0x0" ::: "memory");
  #endif
#endif
        __syncthreads();   // chunk landed; prior readers of buf^1 done

        if (ch + 1 < NCHK)
            stage(ch + 1, buf ^ 1);   // overlap next chunk with compute

        const int k0 = ch * KC;
        #pragma unroll
        for (int kk = 0; kk < KC; kk += 4) {
            // B fragments (4x16 f32): VGPR v holds K = half*2 + v, N = lane%16
            v2f bf[4];
            #pragma unroll
            for (int nt = 0; nt < 4; ++nt) {
                bf[nt][0] = lds_w[buf][kk + half * 2 + 0][nt * 16 + l16][wave];
                bf[nt][1] = lds_w[buf][kk + half * 2 + 1][nt * 16 + l16][wave];
            }

            // A fragments (16x4 f32): M = lane%16 (+16/tile), K = half*2 + v
            const int kgb = k0 + kk + half * 2;          // even -> b64 aligned
            const v2i ee  = *(const v2i*)&tab[kgb];      // ds_load_b64
            v2f af[2];
            #pragma unroll
            for (int v = 0; v < 2; ++v) {
                const int e   = ee[v];
                const int tap = e & 15;
                const int dof = e >> 4;                  // dOff + 65
                const bool ok = (vmask >> tap) & 1;
                #pragma unroll
                for (int mt = 0; mt < 2; ++mt)
                    af[mt][v] = ok ? abase[mt][dof] : 0.0f;
            }

            #pragma unroll
            for (int mt = 0; mt < 2; ++mt)
                #pragma unroll
                for (int nt = 0; nt < 4; ++nt)
                    acc[mt][nt] = __builtin_amdgcn_wmma_f32_16x16x4_f32(
                        false, af[mt], false, bf[nt],
                        (short)0, acc[mt][nt], false, false);
        }
    }

    // ---- epilogue: bias add + store ----
    // C/D layout: VGPR r -> M = r + 8*half, N = lane%16
    #pragma unroll
    for (int nt = 0; nt < 4; ++nt) {
        const int o = nt * 16 + l16;
        const float bv = bias[(size_t)o * LTOT + p];
        #pragma unroll
        for (int mt = 0; mt < 2; ++mt) {
            #pragma unroll
            for (int r = 0; r < 8; ++r) {
                const int b = mt * 16 + half * 8 + r;
                out[((size_t)b * CO + o) * LTOT + p] = acc[mt][nt][r] + bv;
            }
        }
    }
}

extern "C" void kernel_launch(void* const* d_in, const int* in_sizes, int n_in,
                              void* d_out, int out_size, void* d_ws, size_t ws_size,
                              hipStream_t stream) {
    const float* x = (const float*)d_in[0];   // 32*32*64*64
    const float* w = (const float*)d_in[1];   // 64*288*4096
    const float* b = (const float*)d_in[2];   // 64*4096
    float* out = (float*)d_out;               // 32*64*4096

    (void)in_sizes; (void)n_in; (void)out_size; (void)d_ws; (void)ws_size;

    dim3 grid(LTOT / PBLK);   // 512 workgroups
    dim3 block(256);          // 8 waves, one spatial location each
    rrf2d_wmma_kernel<<<grid, block, 0, stream>>>(x, w, b, out);
}